// ProductManifoldAttention_6390911336726
// MI455X (gfx1250) — compile-verified
//
#include <hip/hip_runtime.h>
#include <math.h>

// Problem constants (from reference setup_inputs)
#define B_  8
#define M_  512      // N*L
#define A_  64
#define DE_ 128
#define DH_ 64
#define ZD_ 128
#define ZE_ 64
#define ZH_ 64

#define MAXN (1.0f - 1e-5f)   // (1-eps)/sqrt(c), c=1

typedef float v2f __attribute__((ext_vector_type(2)));
typedef float v8f __attribute__((ext_vector_type(8)));

__device__ __forceinline__ float waveAllSum(float v) {
    v += __shfl_xor(v, 16, 32);
    v += __shfl_xor(v, 8, 32);
    v += __shfl_xor(v, 4, 32);
    v += __shfl_xor(v, 2, 32);
    v += __shfl_xor(v, 1, 32);
    return v;
}

// block-wide sum of per-thread values (threads that don't contribute pass 0)
__device__ __forceinline__ float blockSum(float v, float* red, int tid) {
    red[tid] = v;
    __syncthreads();
    if (tid == 0) {
        float s = 0.f;
        #pragma unroll 8
        for (int i = 0; i < 128; ++i) s += red[i];
        red[128] = s;
    }
    __syncthreads();
    float s = red[128];
    __syncthreads();
    return s;
}

// ---------------------------------------------------------------------------
// Kernel 1: per (b,m) hyperbolic precompute: dH2[b,m] and log_map vectors.
// One wave per (b,m); dh=64 -> 2 floats per lane; shuffle reductions.
// ---------------------------------------------------------------------------
__global__ __launch_bounds__(128) void hyp_precompute(
    const float* __restrict__ curr_hyp,   // [B, DH]
    const float* __restrict__ demo_hyp,   // [B*M, DH]
    float* __restrict__ dH2_out,          // [B*M]
    float* __restrict__ logv_out)         // [B*M, DH]
{
    const int tid  = threadIdx.x;
    const int wave = tid >> 5;
    const int lane = tid & 31;
    const int idx  = blockIdx.x * 4 + wave;   // (b*M + m), grid sized exactly
    const int b    = idx / M_;

    float2 x = ((const float2*)(curr_hyp + (size_t)b * DH_))[lane];
    float2 y = ((const float2*)(demo_hyp + (size_t)idx * DH_))[lane];

    // project x to ball
    float x2r = waveAllSum(x.x * x.x + x.y * x.y);
    float xn  = fmaxf(sqrtf(x2r), 1e-15f);
    float xs  = (xn > MAXN) ? (MAXN / xn) : 1.0f;
    x.x *= xs; x.y *= xs;
    float x2 = x2r * xs * xs;
    // project y to ball
    float y2r = waveAllSum(y.x * y.x + y.y * y.y);
    float yn  = fmaxf(sqrtf(y2r), 1e-15f);
    float ys  = (yn > MAXN) ? (MAXN / yn) : 1.0f;
    y.x *= ys; y.y *= ys;
    float y2 = y2r * ys * ys;

    // u = mobius_add(-x, y), c = 1
    float ax = -x.x, ay = -x.y;
    float xy = waveAllSum(ax * y.x + ay * y.y);
    float ca  = 1.0f + 2.0f * xy + y2;
    float cb  = 1.0f - x2;
    float den = fmaxf(1.0f + 2.0f * xy + x2 * y2, 1e-15f);
    float ux = (ca * ax + cb * y.x) / den;
    float uy = (ca * ay + cb * y.y) / den;

    float u2 = waveAllSum(ux * ux + uy * uy);
    float un = fmaxf(sqrtf(u2), 1e-15f);
    float at = atanhf(fminf(un, 1.0f - 1e-7f));

    float dist = 2.0f * at;                            // sqrt_c = 1
    float lam  = 2.0f / fmaxf(1.0f - x2, 1e-15f);
    float scl  = (2.0f / lam) * at / un;               // log-map scale / unorm

    float2 lv = { scl * ux, scl * uy };
    ((float2*)(logv_out + (size_t)idx * DH_))[lane] = lv;
    if (lane == 0) dH2_out[idx] = dist * dist;
}

// ---------------------------------------------------------------------------
// Kernel 2: flash-style attention, one 128-thread block per (b,a).
// Single pass over demo_rho (134 MB total, each byte read once): online
// softmax with fused Euclidean + hyperbolic aggregation, then exp-map.
// ---------------------------------------------------------------------------
__global__ __launch_bounds__(128) void attention_main(
    const float* __restrict__ curr_rho,   // [B, A, DE]
    const float* __restrict__ curr_hyp,   // [B, DH]
    const float* __restrict__ demo_rho,   // [B, M, A, DE]
    const float* __restrict__ dH2_ws,     // [B*M]
    const float* __restrict__ logv_ws,    // [B*M, DH]
    float* __restrict__ eout_ws,          // [B*A, DE]
    float* __restrict__ hout_ws)          // [B*A, DH]
{
    const int tid  = threadIdx.x;
    const int wave = tid >> 5;
    const int lane = tid & 31;
    const int b = blockIdx.x / A_;
    const int a = blockIdx.x % A_;

    __shared__ float s_mx[4], s_Z[4];
    __shared__ float s_acce[4][DE_];
    __shared__ float s_accv[4][DH_];
    __shared__ float s_red[129];

    const float4 xv = ((const float4*)(curr_rho + ((size_t)b * A_ + a) * DE_))[lane];

    float  mx = -INFINITY, Z = 0.f;
    float4 ae = {0.f, 0.f, 0.f, 0.f};
    float2 av = {0.f, 0.f};

    for (int m = wave; m < M_; m += 4) {
        const size_t bm = (size_t)b * M_ + m;
        float4 r  = ((const float4*)(demo_rho + (bm * A_ + a) * (size_t)DE_))[lane];
        float2 lv = ((const float2*)(logv_ws + bm * DH_))[lane];

        float dx = r.x - xv.x, dy = r.y - xv.y, dz = r.z - xv.z, dw = r.w - xv.w;
        float e2 = waveAllSum(dx*dx + dy*dy + dz*dz + dw*dw);
        float s  = -(dH2_ws[bm] + e2);        // LH=LE=TAU=1

        float nm = fmaxf(mx, s);
        float sc = __expf(mx - nm);           // first iter: exp(-inf)=0
        float p  = __expf(s - nm);
        Z = Z * sc + p;
        ae.x = ae.x * sc + p * r.x;
        ae.y = ae.y * sc + p * r.y;
        ae.z = ae.z * sc + p * r.z;
        ae.w = ae.w * sc + p * r.w;
        av.x = av.x * sc + p * lv.x;
        av.y = av.y * sc + p * lv.y;
        mx = nm;
    }

    s_acce[wave][lane * 4 + 0] = ae.x;
    s_acce[wave][lane * 4 + 1] = ae.y;
    s_acce[wave][lane * 4 + 2] = ae.z;
    s_acce[wave][lane * 4 + 3] = ae.w;
    s_accv[wave][lane * 2 + 0] = av.x;
    s_accv[wave][lane * 2 + 1] = av.y;
    if (lane == 0) { s_mx[wave] = mx; s_Z[wave] = Z; }
    __syncthreads();

    // merge 4 waves' online-softmax partials
    float gm = fmaxf(fmaxf(s_mx[0], s_mx[1]), fmaxf(s_mx[2], s_mx[3]));
    float w0 = __expf(s_mx[0] - gm), w1 = __expf(s_mx[1] - gm);
    float w2 = __expf(s_mx[2] - gm), w3 = __expf(s_mx[3] - gm);
    float invZ = 1.0f / (w0*s_Z[0] + w1*s_Z[1] + w2*s_Z[2] + w3*s_Z[3]);

    float e_out = (w0*s_acce[0][tid] + w1*s_acce[1][tid] +
                   w2*s_acce[2][tid] + w3*s_acce[3][tid]) * invZ;
    float vd = 0.f;
    if (tid < DH_)
        vd = (w0*s_accv[0][tid] + w1*s_accv[1][tid] +
              w2*s_accv[2][tid] + w3*s_accv[3][tid]) * invZ;
    __syncthreads();

    // exp-map: h_out = exp_map_x(curr_hyp[b], v), c = 1
    float xd = (tid < DH_) ? curr_hyp[(size_t)b * DH_ + tid] : 0.f;
    float x2r = blockSum(xd * xd, s_red, tid);
    float xn  = fmaxf(sqrtf(x2r), 1e-15f);
    float xs  = (xn > MAXN) ? (MAXN / xn) : 1.f;
    xd *= xs;
    float x2  = x2r * xs * xs;
    float lam = 2.f / fmaxf(1.f - x2, 1e-15f);

    float v2 = blockSum(vd * vd, s_red, tid);
    float vn = fmaxf(sqrtf(v2), 1e-15f);
    float factor = tanhf(lam * vn * 0.5f);
    float yd = (vd / vn) * factor;

    float y2r = blockSum(yd * yd, s_red, tid);
    float ynn = fmaxf(sqrtf(y2r), 1e-15f);
    float ysc = (ynn > MAXN) ? (MAXN / ynn) : 1.f;
    yd *= ysc;
    float y2 = y2r * ysc * ysc;

    float xyv = blockSum(xd * yd, s_red, tid);
    float den = fmaxf(1.f + 2.f*xyv + x2*y2, 1e-15f);
    float od  = ((1.f + 2.f*xyv + y2) * xd + (1.f - x2) * yd) / den;

    float o2  = blockSum(od * od, s_red, tid);
    float onn = fmaxf(sqrtf(o2), 1e-15f);
    float osc = (onn > MAXN) ? (MAXN / onn) : 1.f;
    od *= osc;

    eout_ws[(size_t)blockIdx.x * DE_ + tid] = e_out;
    if (tid < DH_) hout_ws[(size_t)blockIdx.x * DH_ + tid] = od;
}

// ---------------------------------------------------------------------------
// Kernel 3: fused projection GEMMs (fp32 WMMA 16x16x4) + LayerNorm.
// 256 threads = 8 waves: waves 0-3 -> e_out @ We.T tile-cols, 4-7 -> h_out @ Wh.T.
// ---------------------------------------------------------------------------
__global__ __launch_bounds__(256) void proj_ln_wmma(
    const float* __restrict__ eout_ws,    // [B*A, DE]
    const float* __restrict__ hout_ws,    // [B*A, DH]
    const float* __restrict__ We,         // [ZE, DE]
    const float* __restrict__ Wh,         // [ZH, DH]
    const float* __restrict__ gamma,      // [ZD]
    const float* __restrict__ beta,       // [ZD]
    float* __restrict__ out)              // [B*A, ZD]
{
    __shared__ float s_et[16][DE_];
    __shared__ float s_ht[16][DH_];
    __shared__ float s_z[16][ZD_];
    __shared__ float s_s1[16][16];
    __shared__ float s_s2[16][16];
    __shared__ float s_mean[16], s_rstd[16];

    const int tid  = threadIdx.x;
    const int wave = tid >> 5;
    const int lane = tid & 31;
    const int half = lane >> 4;
    const int l16  = lane & 15;
    const int r0   = blockIdx.x * 16;

    for (int i = tid; i < 16 * DE_; i += 256)
        s_et[i >> 7][i & 127] = eout_ws[(size_t)r0 * DE_ + i];
    for (int i = tid; i < 16 * DH_; i += 256)
        s_ht[i >> 6][i & 63] = hout_ws[(size_t)r0 * DH_ + i];
    __syncthreads();

    v8f c = {0.f, 0.f, 0.f, 0.f, 0.f, 0.f, 0.f, 0.f};
    if (wave < 4) {                       // wave-uniform branch: EXEC all-ones
        const int j0 = wave * 16;
        for (int kk = 0; kk < DE_ / 4; ++kk) {
            const int kb = kk * 4 + (half ? 2 : 0);
            v2f af, bf;
            af[0] = s_et[l16][kb];
            af[1] = s_et[l16][kb + 1];
            bf[0] = We[(j0 + l16) * DE_ + kb];
            bf[1] = We[(j0 + l16) * DE_ + kb + 1];
            c = __builtin_amdgcn_wmma_f32_16x16x4_f32(
                    false, af, false, bf, (short)0, c, false, false);
        }
        #pragma unroll
        for (int v = 0; v < 8; ++v)
            s_z[v + 8 * half][j0 + l16] = c[v];
    } else {
        const int j0 = (wave - 4) * 16;
        for (int kk = 0; kk < DH_ / 4; ++kk) {
            const int kb = kk * 4 + (half ? 2 : 0);
            v2f af, bf;
            af[0] = s_ht[l16][kb];
            af[1] = s_ht[l16][kb + 1];
            bf[0] = Wh[(j0 + l16) * DH_ + kb];
            bf[1] = Wh[(j0 + l16) * DH_ + kb + 1];
            c = __builtin_amdgcn_wmma_f32_16x16x4_f32(
                    false, af, false, bf, (short)0, c, false, false);
        }
        #pragma unroll
        for (int v = 0; v < 8; ++v)
            s_z[v + 8 * half][ZE_ + j0 + l16] = c[v];
    }
    __syncthreads();

    // LayerNorm over ZD=128 per row; 16 threads per row, 8 cols each.
    const int row = tid >> 4, seg = tid & 15;
    float s1 = 0.f, s2 = 0.f;
    #pragma unroll
    for (int i = 0; i < 8; ++i) {
        float z = s_z[row][seg * 8 + i];
        s1 += z; s2 += z * z;
    }
    s_s1[row][seg] = s1;
    s_s2[row][seg] = s2;
    __syncthreads();
    if (seg == 0) {
        float a1 = 0.f, a2 = 0.f;
        #pragma unroll
        for (int i = 0; i < 16; ++i) { a1 += s_s1[row][i]; a2 += s_s2[row][i]; }
        float mean = a1 * (1.0f / ZD_);
        float var  = a2 * (1.0f / ZD_) - mean * mean;
        s_mean[row] = mean;
        s_rstd[row] = rsqrtf(var + 1e-5f);
    }
    __syncthreads();
    const float mean = s_mean[row], rstd = s_rstd[row];
    #pragma unroll
    for (int i = 0; i < 8; ++i) {
        const int col = seg * 8 + i;
        float z = (s_z[row][col] - mean) * rstd * gamma[col] + beta[col];
        out[(size_t)(r0 + row) * ZD_ + col] = z;
    }
}

// ---------------------------------------------------------------------------
extern "C" void kernel_launch(void* const* d_in, const int* in_sizes, int n_in,
                              void* d_out, int out_size, void* d_ws, size_t ws_size,
                              hipStream_t stream) {
    (void)in_sizes; (void)n_in; (void)out_size; (void)ws_size;
    const float* curr_rho = (const float*)d_in[0];   // [B,A,DE]
    const float* curr_hyp = (const float*)d_in[1];   // [B,DH]
    const float* demo_rho = (const float*)d_in[2];   // [B,M,A,DE]
    const float* demo_hyp = (const float*)d_in[3];   // [B,M,DH]
    const float* We       = (const float*)d_in[4];   // [ZE,DE]
    const float* Wh       = (const float*)d_in[5];   // [ZH,DH]
    const float* gamma    = (const float*)d_in[6];   // [ZD]
    const float* beta     = (const float*)d_in[7];   // [ZD]
    float* out = (float*)d_out;

    // workspace carve: dH2 [B*M] | logv [B*M,DH] | e_out [B*A,DE] | h_out [B*A,DH]
    float* ws      = (float*)d_ws;
    float* dH2_ws  = ws;                                   //   4096 floats
    float* logv_ws = dH2_ws + (size_t)B_ * M_;             // 262144 floats
    float* eout_ws = logv_ws + (size_t)B_ * M_ * DH_;      //  65536 floats
    float* hout_ws = eout_ws + (size_t)B_ * A_ * DE_;      //  32768 floats

    hyp_precompute<<<(B_ * M_) / 4, 128, 0, stream>>>(curr_hyp, demo_hyp,
                                                      dH2_ws, logv_ws);
    attention_main<<<B_ * A_, 128, 0, stream>>>(curr_rho, curr_hyp, demo_rho,
                                                dH2_ws, logv_ws, eout_ws, hout_ws);
    proj_ln_wmma<<<(B_ * A_) / 16, 256, 0, stream>>>(eout_ws, hout_ws, We, Wh,
                                                     gamma, beta, out);
}